// HierarchicalSparseAttentionTriton_17549236371568
// MI455X (gfx1250) — compile-verified
//
#include <hip/hip_runtime.h>

typedef float v2f __attribute__((ext_vector_type(2)));
typedef float v8f __attribute__((ext_vector_type(8)));

#define B_DIM 4
#define S_DIM 4096
#define H_DIM 8
#define D_DIM 64
#define SCALE 0.125f
#define NTILE (S_DIM / 16)   // 256 query tiles per (b,h)

// ---------------------------------------------------------------------------
// Tree build: level `level` parents from level-1 children (softmax merge).
// Internal node storage Kt/Vt: (B, H, 4096 nodes, 64), node row for tree level
// l (1..11) starts at IB(l) = S - (S >> (l-1)).
// One wave per parent; lane owns 2 of the 64 dims.
// ---------------------------------------------------------------------------
__global__ void __launch_bounds__(256)
tree_level_kernel(const float* __restrict__ kleaf,
                  const float* __restrict__ vleaf,
                  float* __restrict__ Kt,
                  float* __restrict__ Vt,
                  int level, int P)
{
    const int wave = threadIdx.x >> 5;
    const int lane = threadIdx.x & 31;
    const long g = (long)blockIdx.x * 8 + wave;
    const long total = (long)B_DIM * H_DIM * P;
    if (g >= total) return;
    const int p  = (int)(g % P);
    const int bh = (int)(g / P);
    const int h  = bh % H_DIM;
    const int b  = bh / H_DIM;

    const int doff = 2 * lane;
    float2 kc0, kc1, vc0, vc1;
    if (level == 1) {
        const size_t r0 = ((size_t)(b * S_DIM + 2 * p    ) * H_DIM + h) * D_DIM;
        const size_t r1 = ((size_t)(b * S_DIM + 2 * p + 1) * H_DIM + h) * D_DIM;
        kc0 = *(const float2*)(kleaf + r0 + doff);
        kc1 = *(const float2*)(kleaf + r1 + doff);
        vc0 = *(const float2*)(vleaf + r0 + doff);
        vc1 = *(const float2*)(vleaf + r1 + doff);
    } else {
        const int IBc = S_DIM - (S_DIM >> (level - 2));      // child level base
        const size_t tb = (size_t)(b * H_DIM + h) * S_DIM;
        const size_t r0 = (tb + IBc + 2 * p    ) * D_DIM;
        const size_t r1 = (tb + IBc + 2 * p + 1) * D_DIM;
        kc0 = *(const float2*)(Kt + r0 + doff);
        kc1 = *(const float2*)(Kt + r1 + doff);
        vc0 = *(const float2*)(Vt + r0 + doff);
        vc1 = *(const float2*)(Vt + r1 + doff);
    }

    float2 kp, vp;
    kp.x = 0.5f * (kc0.x + kc1.x); kp.y = 0.5f * (kc0.y + kc1.y);
    vp.x = 0.5f * (vc0.x + vc1.x); vp.y = 0.5f * (vc0.y + vc1.y);

    float pss = kp.x * kp.x  + kp.y * kp.y;
    float ps0 = kp.x * kc0.x + kp.y * kc0.y;
    float ps1 = kp.x * kc1.x + kp.y * kc1.y;
    #pragma unroll
    for (int m = 16; m >= 1; m >>= 1) {
        pss += __shfl_xor(pss, m, 32);
        ps0 += __shfl_xor(ps0, m, 32);
        ps1 += __shfl_xor(ps1, m, 32);
    }
    pss *= SCALE; ps0 *= SCALE; ps1 *= SCALE;
    const float mm = fmaxf(pss, fmaxf(ps0, ps1));
    const float es = __expf(pss - mm);
    const float e0 = __expf(ps0 - mm);
    const float e1 = __expf(ps1 - mm);
    const float inv = 1.0f / (es + e0 + e1 + 1e-9f);

    float2 vo;
    vo.x = (es * vp.x + e0 * vc0.x + e1 * vc1.x) * inv;
    vo.y = (es * vp.y + e0 * vc0.y + e1 * vc1.y) * inv;

    const int IBl = S_DIM - (S_DIM >> (level - 1));
    const size_t dst = ((size_t)(b * H_DIM + h) * S_DIM + IBl + p) * D_DIM + doff;
    *(float2*)(Kt + dst) = kp;
    *(float2*)(Vt + dst) = vo;
}

// ---------------------------------------------------------------------------
// Attention: one wave per 16-query tile. Candidate keys per tile packed into
// three 16-key groups; scores and E*V both via V_WMMA_F32_16X16X4_F32.
// ---------------------------------------------------------------------------
__global__ void __launch_bounds__(128)
hst_attn_kernel(const float* __restrict__ q,
                const float* __restrict__ k,
                const float* __restrict__ v,
                const float* __restrict__ Kt,
                const float* __restrict__ Vt,
                float* __restrict__ out)
{
    __shared__ float ldsS[4][784];   // per wave: 3x(16x16) scores/E + 16 denom
    __shared__ int   ldsRow[4][32];  // per wave: rowB[16], rowC[16]

    const int lane = threadIdx.x & 31;
    const int w    = threadIdx.x >> 5;
    const int tile = blockIdx.x * 4 + w;
    const int t  = tile & (NTILE - 1);
    const int bh = tile >> 8;            // / NTILE
    const int h  = bh & (H_DIM - 1);
    const int b  = bh >> 3;

    const int n    = lane & 15;
    const int half = lane >> 4;

    float* sA  = &ldsS[w][0];
    float* sB  = &ldsS[w][256];
    float* sC  = &ldsS[w][512];
    float* den = &ldsS[w][768];
    int*   rowB = &ldsRow[w][0];
    int*   rowC = &ldsRow[w][16];

    // Internal-node rows for candidate slot n of groups B and C.
    int rB;
    if      (n <  8) rB = 8 * t + n;                            // lvl1: IB=0
    else if (n < 12) rB = 2048 + 4 * t + (n - 8);               // lvl2
    else if (n < 14) rB = 3072 + 2 * t + (n - 12);              // lvl3
    else if (n == 14) rB = 3584 + (t ^ 1);                      // lvl4 sibling
    else              rB = 3840 + ((t >> 1) ^ 1);               // lvl5 sibling
    const int rC = (n < 6) ? (S_DIM - (S_DIM >> (5 + n))) + ((t >> (n + 2)) ^ 1)
                           : 0;                                 // lvl 6+n sibling
    if (half == 0) { rowB[n] = rB; rowC[n] = rC; }

    const size_t leafBase = ((size_t)(b * S_DIM + 16 * t + n) * H_DIM + h) * D_DIM;
    const size_t treeBase = (size_t)(b * H_DIM + h) * S_DIM * (size_t)D_DIM;
    const float* qp  = q  + leafBase;
    const float* kAp = k  + leafBase;
    const float* kBp = Kt + treeBase + (size_t)rB * D_DIM;
    const float* kCp = Kt + treeBase + (size_t)rC * D_DIM;

    // ---- scores: 3 accumulators, 16 chained K=4 f32 WMMAs each ----
    v8f cA = {}; v8f cB = {}; v8f cC = {};
    const int dofs = 2 * half;
    #pragma unroll
    for (int ks = 0; ks < 16; ++ks) {
        const int d = 4 * ks + dofs;
        v2f a  = *(const v2f*)(qp  + d);
        a.x *= SCALE; a.y *= SCALE;
        v2f bA = *(const v2f*)(kAp + d);
        v2f bB = *(const v2f*)(kBp + d);
        v2f bC = *(const v2f*)(kCp + d);
        cA = __builtin_amdgcn_wmma_f32_16x16x4_f32(false, a, false, bA, (short)0, cA, false, false);
        cB = __builtin_amdgcn_wmma_f32_16x16x4_f32(false, a, false, bB, (short)0, cB, false, false);
        cC = __builtin_amdgcn_wmma_f32_16x16x4_f32(false, a, false, bC, (short)0, cC, false, false);
    }

    #pragma unroll
    for (int r = 0; r < 8; ++r) {
        const int M = r + 8 * half;
        sA[M * 16 + n] = cA[r];
        sB[M * 16 + n] = cB[r];
        sC[M * 16 + n] = cC[r];
    }
    __syncthreads();

    // ---- per-query softmax over 13 candidates (lane j<16 owns query j) ----
    float ev[13];
    float dsum = 0.0f;
    if (half == 0) {
        const int j = n;
        float x[13];
        x[0] = sA[j * 16 + j];                         // self
        x[1] = sA[j * 16 + (j ^ 1)];                   // lvl0
        x[2] = sB[j * 16 + ((j >> 1) ^ 1)];            // lvl1
        x[3] = sB[j * 16 + 8  + ((j >> 2) ^ 1)];       // lvl2
        x[4] = sB[j * 16 + 12 + ((j >> 3) ^ 1)];       // lvl3
        x[5] = sB[j * 16 + 14];                        // lvl4
        x[6] = sB[j * 16 + 15];                        // lvl5
        #pragma unroll
        for (int i = 0; i < 6; ++i) x[7 + i] = sC[j * 16 + i];   // lvl6..11

        bool msk[13];
        msk[0] = false;
        #pragma unroll
        for (int l = 0; l < 12; ++l) {
            const int bit = (l < 4) ? ((j >> l) & 1) : ((t >> (l - 4)) & 1);
            msk[l + 1] = (bit == 0);                   // even node => masked
        }
        float m = x[0];
        #pragma unroll
        for (int i = 1; i < 13; ++i) if (!msk[i]) m = fmaxf(m, x[i]);
        #pragma unroll
        for (int i = 0; i < 13; ++i) {
            ev[i] = msk[i] ? 0.0f : __expf(x[i] - m);
            dsum += ev[i];
        }
    }
    __syncthreads();

    // ---- rebuild sparse E matrices over the score buffers ----
    #pragma unroll
    for (int i = lane; i < 768; i += 32) ldsS[w][i] = 0.0f;
    __syncthreads();
    if (half == 0) {
        const int j = n;
        den[j] = dsum;
        sA[j * 16 + j]                   = ev[0];
        sA[j * 16 + (j ^ 1)]             = ev[1];
        sB[j * 16 + ((j >> 1) ^ 1)]      = ev[2];
        sB[j * 16 + 8  + ((j >> 2) ^ 1)] = ev[3];
        sB[j * 16 + 12 + ((j >> 3) ^ 1)] = ev[4];
        sB[j * 16 + 14]                  = ev[5];
        sB[j * 16 + 15]                  = ev[6];
        #pragma unroll
        for (int i = 0; i < 6; ++i) sC[j * 16 + i] = ev[7 + i];
    }
    __syncthreads();

    // ---- output: O = EA*VA + EB*VB + EC*VC (16x64), K=16 per group ----
    const size_t vLeafRow0  = ((size_t)(b * S_DIM + 16 * t) * H_DIM + h) * (size_t)D_DIM;
    const size_t vRowStride = (size_t)H_DIM * D_DIM;    // 512 floats
    #pragma unroll
    for (int dsl = 0; dsl < 4; ++dsl) {
        const int dcol = 16 * dsl + n;
        v8f acc = {};
        #pragma unroll
        for (int ks2 = 0; ks2 < 4; ++ks2) {
            const int c0 = 4 * ks2 + 2 * half;          // this lane's 2 cand rows
            v2f aE, bV;
            // group A: leaf values
            aE.x = sA[n * 16 + c0]; aE.y = sA[n * 16 + c0 + 1];
            bV.x = v[vLeafRow0 + (size_t)c0       * vRowStride + dcol];
            bV.y = v[vLeafRow0 + (size_t)(c0 + 1) * vRowStride + dcol];
            acc = __builtin_amdgcn_wmma_f32_16x16x4_f32(false, aE, false, bV, (short)0, acc, false, false);
            // group B: internal nodes lvl1..5
            aE.x = sB[n * 16 + c0]; aE.y = sB[n * 16 + c0 + 1];
            bV.x = Vt[treeBase + (size_t)rowB[c0]     * D_DIM + dcol];
            bV.y = Vt[treeBase + (size_t)rowB[c0 + 1] * D_DIM + dcol];
            acc = __builtin_amdgcn_wmma_f32_16x16x4_f32(false, aE, false, bV, (short)0, acc, false, false);
            // group C: internal nodes lvl6..11 (padded slots have E=0)
            aE.x = sC[n * 16 + c0]; aE.y = sC[n * 16 + c0 + 1];
            bV.x = Vt[treeBase + (size_t)rowC[c0]     * D_DIM + dcol];
            bV.y = Vt[treeBase + (size_t)rowC[c0 + 1] * D_DIM + dcol];
            acc = __builtin_amdgcn_wmma_f32_16x16x4_f32(false, aE, false, bV, (short)0, acc, false, false);
        }
        #pragma unroll
        for (int r = 0; r < 8; ++r) {
            const int M = r + 8 * half;
            out[((size_t)(b * S_DIM + 16 * t + M) * H_DIM + h) * D_DIM + dcol] =
                acc[r] / den[M];
        }
    }
}

// ---------------------------------------------------------------------------
extern "C" void kernel_launch(void* const* d_in, const int* in_sizes, int n_in,
                              void* d_out, int out_size, void* d_ws, size_t ws_size,
                              hipStream_t stream)
{
    const float* q = (const float*)d_in[0];
    const float* k = (const float*)d_in[1];
    const float* v = (const float*)d_in[2];
    float* out = (float*)d_out;

    // Workspace: Kt then Vt, each B*H*4096*64 floats (32 MiB) -> 64 MiB total.
    float* Kt = (float*)d_ws;
    float* Vt = Kt + (size_t)B_DIM * H_DIM * S_DIM * D_DIM;

    for (int level = 1; level <= 11; ++level) {          // root never gathered
        const int P = S_DIM >> level;
        const long total = (long)B_DIM * H_DIM * P;
        const int blocks = (int)((total + 7) / 8);       // 8 waves / block
        tree_level_kernel<<<blocks, 256, 0, stream>>>(k, v, Kt, Vt, level, P);
    }

    const int tiles = B_DIM * H_DIM * NTILE;             // 8192, /4 waves per block
    hst_attn_kernel<<<tiles / 4, 128, 0, stream>>>(q, k, v, Kt, Vt, out);

    (void)in_sizes; (void)n_in; (void)out_size; (void)ws_size;
}